// OrientationFilter_28793460752640
// MI455X (gfx1250) — compile-verified
//
#include <hip/hip_runtime.h>
#include <stdint.h>

// Per-element 2-state Kalman filter (orientation): streaming, memory-bound.
// Traffic: 52 B in + 24 B out per element x 4M = 304 MB -> ~13 us floor at
// 23.3 TB/s. Compute ~50 flops/elem (0.66 flop/byte) -> HBM-limited; WMMA is
// inapplicable (2x2 per-element matrices). CDNA5 features that matter here:
//  - async global->LDS b128 copies (ASYNCcnt) for the 16B/lane streams
//  - non-temporal hints everywhere (dataset > 192MB L2, single-use data)
//  - wave32 sizing: 256-thread blocks = 8 waves, 15625 blocks in flight.

typedef __attribute__((ext_vector_type(4))) float v4f;
typedef __attribute__((ext_vector_type(2))) float v2f;

#define TWO_PI_F     6.28318530717958647692f
#define INV_TWO_PI_F 0.15915494309189533577f

__device__ __forceinline__ float wrap_pi(float a) {
    // a - 2*pi*round(a/2*pi); rintf = round-to-nearest-even (matches jnp.round)
    return __builtin_fmaf(-TWO_PI_F, rintf(a * INV_TWO_PI_F), a);
}

__global__ __launch_bounds__(256) void kf_orientation_kernel(
    const float* __restrict__ x0_g,   // [B,2,1]
    const float* __restrict__ P0_g,   // [B,2,2]
    const float* __restrict__ dt_g,   // [B]
    const float* __restrict__ Q_g,    // [B,2,2] (diagonal)
    const float* __restrict__ z_g,    // [B,1,1]
    const float* __restrict__ R_g,    // [B,1,1]
    float* __restrict__ out_x,        // [B,2,1]
    float* __restrict__ out_P,        // [B,2,2]
    int n)
{
    __shared__ v4f sP0[256];
    __shared__ v4f sQ [256];

    const int tid = threadIdx.x;
    const int i   = blockIdx.x * 256 + tid;
    if (i >= n) return;

    // ---- stage the two 16B/lane streams through LDS with CDNA5 async copies
    const v4f* gP0 = (const v4f*)P0_g + i;
    const v4f* gQ  = (const v4f*)Q_g  + i;

    // LDS byte offsets: generic LDS addresses carry the wg-relative offset in
    // the low 32 bits (aperture base lives in bits [63:32]).
    unsigned lP0 = (unsigned)(size_t)&sP0[tid];
    unsigned lQ  = (unsigned)(size_t)&sQ [tid];

    asm volatile("global_load_async_to_lds_b128 %0, %1, off"
                 :: "v"(lP0), "v"(gP0) : "memory");
    asm volatile("global_load_async_to_lds_b128 %0, %1, off"
                 :: "v"(lQ),  "v"(gQ)  : "memory");

    // ---- overlap: remaining streams via direct non-temporal loads ----
    const v2f  x  = __builtin_nontemporal_load((const v2f*)x0_g + i); // (theta, omega)
    const float dt = __builtin_nontemporal_load(dt_g + i);
    const float z  = __builtin_nontemporal_load(z_g  + i);
    const float R  = __builtin_nontemporal_load(R_g  + i);

    // Drain async copies (each lane reads only what it requested: no barrier).
    asm volatile("s_wait_asynccnt 0x0" ::: "memory");

    const v4f p = sP0[tid];   // [[p.x p.y],[p.z p.w]]
    const v4f q = sQ [tid];   // diag(q.x, q.w); q.y = q.z = 0 in memory

    // ---- predict: x = F x0, F = [[1,dt],[0,1]] ----
    float theta = wrap_pi(x.x + dt * x.y);
    float omega = x.y;

    // P = F P0 F^T + Q
    // A = F*P0 = [[p00+dt*p10, p01+dt*p11],[p10, p11]]
    const float a00 = p.x + dt * p.z;
    const float a01 = p.y + dt * p.w;
    const float a10 = p.z;
    const float a11 = p.w;
    // (A F^T)[i][0] = a_i0 + dt*a_i1 ; (A F^T)[i][1] = a_i1
    const float P00 = a00 + dt * a01 + q.x;
    const float P01 = a01 + q.y;
    const float P10 = a10 + dt * a11 + q.z;
    const float P11 = a11 + q.w;

    // ---- update: H = [1 0] ----
    const float S  = P00 + R;
    const float K0 = P00 / S;       // P H^T / S
    const float K1 = P10 / S;
    const float innov = wrap_pi(z - theta);

    const float xn0 = wrap_pi(theta + K0 * innov);
    const float xn1 = omega + K1 * innov;

    // P = (I - K H) P ; (I-KH) = [[1-K0, 0], [-K1, 1]]
    const float omk = 1.0f - K0;
    const float Pn00 = omk * P00;
    const float Pn01 = omk * P01;
    const float Pn10 = P10 - K1 * P00;
    const float Pn11 = P11 - K1 * P01;

    // ---- non-temporal, coalesced stores (single-use data, > L2 footprint) ----
    v2f xo; xo.x = xn0; xo.y = xn1;
    v4f Po; Po.x = Pn00; Po.y = Pn01; Po.z = Pn10; Po.w = Pn11;
    __builtin_nontemporal_store(xo, (v2f*)out_x + i);
    __builtin_nontemporal_store(Po, (v4f*)out_P + i);
}

extern "C" void kernel_launch(void* const* d_in, const int* in_sizes, int n_in,
                              void* d_out, int out_size, void* d_ws, size_t ws_size,
                              hipStream_t stream) {
    // setup_inputs order: x0, P0, delta_t, Q, z, R
    const float* x0 = (const float*)d_in[0];
    const float* P0 = (const float*)d_in[1];
    const float* dt = (const float*)d_in[2];
    const float* Q  = (const float*)d_in[3];
    const float* z  = (const float*)d_in[4];
    const float* R  = (const float*)d_in[5];
    const int n = in_sizes[2];            // B (delta_t is [B])
    if (n <= 0) return;

    float* out_x = (float*)d_out;                 // first tuple member: x [B,2,1]
    float* out_P = out_x + (size_t)2 * (size_t)n; // second: P [B,2,2]

    const int blocks = (n + 255) / 256;           // B = 4M -> 15625 full blocks
    kf_orientation_kernel<<<blocks, 256, 0, stream>>>(x0, P0, dt, Q, z, R,
                                                      out_x, out_P, n);
}